// FloodFeatureGraph_23759759081723
// MI455X (gfx1250) — compile-verified
//
#include <hip/hip_runtime.h>
#include <hip/hip_bf16.h>

typedef float v2f __attribute__((ext_vector_type(2)));
typedef float v8f __attribute__((ext_vector_type(8)));

constexpr int   RBF_N   = 20;
constexpr float INV_STD = 19.0f / 20.0f;   // (D - c*std)/std = D*19/20 - c, std = 20/19
constexpr float L2E     = 1.4426950408889634f;  // log2(e)
constexpr float LN2     = 0.6931471805599453f;  // ln(2)
constexpr int   PAIRS   = 64;                   // P^2, P = 2*A = 8
constexpr int   FEAT    = PAIRS * RBF_N + PAIRS; // 1344 floats per edge

// One wave handles 2 edges (16 points). Two chained V_WMMA_F32_16X16X4_F32
// build the 16x16 squared-distance matrix (|p|^2 + |q|^2 - 2 p.q); the two
// diagonal 8x8 blocks are the edges' D^2. Productive lanes bounce D^2 through
// LDS so ALL 32 lanes featurize (4 pairs each) with raw v_sqrt/v_exp/v_log.
__global__ __launch_bounds__(256) void flood_feature_kernel(
    const float* __restrict__ X,        // (N, 4, 3)
    const int*   __restrict__ edge_idx, // (N, K)
    const int*   __restrict__ C,        // (N,)
    float*       __restrict__ out,      // (E, 1344)
    int K, long long E)
{
    __shared__ float4 s_pts[8][16];    // per wave: 16 points (x,y,z,|p|^2)
    __shared__ float  s_mask[8][2];    // per wave: mask of its 2 edges
    __shared__ float  s_d2[8][128];    // per wave: D^2, layout [edge][col][row]

    const int tid  = threadIdx.x;
    const int wave = tid >> 5;
    const int lane = tid & 31;
    const long long blockBase = (long long)blockIdx.x * 16;

    // ---- stage phase: lanes 0..15 load 2 edges x 8 points into LDS ----
    if (lane < 16) {
        const int s = lane >> 3;          // which edge of this wave
        const int p = lane & 7;           // point 0..7 (0-3: i atoms, 4-7: j atoms)
        long long e  = blockBase + wave * 2 + s;
        long long ec = (e < E) ? e : (E - 1);
        const int n = (int)(ec / K);
        const int k = (int)(ec - (long long)n * K);
        const int j = edge_idx[n * K + k];
        const int src  = (p < 4) ? n : j;
        const float* xp = X + ((size_t)src * 4 + (p & 3)) * 3;
        const float x = xp[0], y = xp[1], z = xp[2];
        s_pts[wave][lane] = make_float4(x, y, z, x * x + y * y + z * z);
        if (p == 0) s_mask[wave][s] = (C[n] > 0 && C[j] > 0) ? 1.0f : 0.0f;
    }
    __syncthreads();

    // ---- WMMA operand build per ISA 32-bit A/B 16x4 layout ----
    // lanes 0-15 carry K=0,1 in v0,v1; lanes 16-31 carry K=2,3.
    const int idx = lane & 15;            // row M (for A) / col N (for B)
    const float4 pt = s_pts[wave][idx];
    v2f a1, b1, a2, b2;
    if (lane < 16) {
        a1[0] = pt.x;          a1[1] = pt.y;          // A1: (x, y, ...)
        b1[0] = -2.0f * pt.x;  b1[1] = -2.0f * pt.y;  // B1: (-2x, -2y, ...)
        a2[0] = pt.w;          a2[1] = 1.0f;          // A2: (|p|^2, 1, 0, 0)
        b2[0] = 1.0f;          b2[1] = pt.w;          // B2: (1, |q|^2, 0, 0)
    } else {
        a1[0] = pt.z;          a1[1] = 0.0f;          // A1: (..., z, 0)
        b1[0] = -2.0f * pt.z;  b1[1] = 0.0f;          // B1: (..., -2z, 0)
        a2[0] = 0.0f;          a2[1] = 0.0f;
        b2[0] = 0.0f;          b2[1] = 0.0f;
    }

    v8f acc = {};
    // acc = A1*B1: -2 * p_m . p_n
    acc = __builtin_amdgcn_wmma_f32_16x16x4_f32(false, a1, false, b1,
                                                (short)0, acc, false, false);
    // acc += A2*B2: + |p_m|^2 + |p_n|^2  ->  acc = D^2[m][n]
    acc = __builtin_amdgcn_wmma_f32_16x16x4_f32(false, a2, false, b2,
                                                (short)0, acc, false, false);

    // ---- spill useful D^2 to LDS (transposed so each lane stores 8 contiguous) ----
    // C/D layout: VGPR r, lanes 0-15 -> (M=r, N=lane); lanes 16-31 -> (M=r+8, N=lane-16).
    // Edge 0 pairs: lanes 0-7 (col=lane). Edge 1 pairs: lanes 24-31 (col=lane-24).
    {
        const bool hi = (lane >= 24);
        if (lane < 8 || hi) {
            const int s   = hi ? 1 : 0;
            const int col = hi ? (lane - 24) : lane;
            float* dst = &s_d2[wave][(s << 6) + (col << 3)]; // [edge][col][row]
            *(float4*)(dst)     = make_float4(acc[0], acc[1], acc[2], acc[3]);
            *(float4*)(dst + 4) = make_float4(acc[4], acc[5], acc[6], acc[7]);
        }
    }
    __syncthreads();

    // ---- featurize + store: all 32 lanes, 4 pairs each (2 edges x 64 pairs) ----
    #pragma unroll
    for (int t = 0; t < 4; ++t) {
        const int q = t * 32 + lane;      // 0..127 across the wave's 2 edges
        const int s = q >> 6;             // edge select
        const int p = q & 63;             // pair index, row-major m*8+n
        const long long e = blockBase + wave * 2 + s;
        if (e < E) {
            const float m  = s_mask[wave][s];
            // stored as [col][row] -> read at n*8 + m (conflict-free per t)
            const float d2 = s_d2[wave][(s << 6) + ((p & 7) << 3) + (p >> 3)];
            const float D  = __builtin_amdgcn_sqrtf(d2 + 1e-6f);   // SQ_EPS
            const float u  = D * INV_STD;
            float* outE    = out + (size_t)e * FEAT;

            float4 buf[5];
            float* bf = (float*)buf;
            #pragma unroll
            for (int c = 0; c < RBF_N; ++c) {
                const float tt = u - (float)c;
                bf[c] = m * __builtin_amdgcn_exp2f(-(tt * tt) * L2E);
            }
            float4* dst = (float4*)(outE + (size_t)p * RBF_N);     // 16B aligned
            #pragma unroll
            for (int v = 0; v < 5; ++v) dst[v] = buf[v];
            // log(D + 0.01) via v_log_f32 (log2) * ln2
            outE[PAIRS * RBF_N + p] = m * (__builtin_amdgcn_logf(D + 0.01f) * LN2);
        }
    }
}

extern "C" void kernel_launch(void* const* d_in, const int* in_sizes, int n_in,
                              void* d_out, int out_size, void* d_ws, size_t ws_size,
                              hipStream_t stream) {
    const float* X        = (const float*)d_in[0];
    const int*   edge_idx = (const int*)d_in[1];
    const int*   C        = (const int*)d_in[2];
    float*       out      = (float*)d_out;

    const int N = in_sizes[2];            // (B=1, N)
    const int K = in_sizes[1] / N;        // (B=1, N, K)
    const long long E = (long long)N * K;

    const int blocks = (int)((E + 15) / 16);   // 16 edges per 256-thread block
    flood_feature_kernel<<<blocks, 256, 0, stream>>>(X, edge_idx, C, out, K, E);
}